// MultiStageFIRFilter_11553462026857
// MI455X (gfx1250) — compile-verified
//
#include <hip/hip_runtime.h>
#include <math.h>

// Problem constants (fixed by the reference setup)
#define T_TOTAL 81920      // N * FRAME_PERIOD
#define NFRAMES 1024
#define MTAPS   24
#define NCOEF   25
#define PERIOD  80
#define STAGES  20
#define HALO    (STAGES * MTAPS)      // 480
#define TILE    544                   // outputs per block
#define SLOTS   (TILE + HALO)         // 1024 time slots per block (16-aligned)
#define THREADS 512                   // 2 slots per thread
#define NTILES  ((T_TOTAL + TILE - 1) / TILE)   // 151
#define CHUNK_SLOTS 256               // slots interpolated per WMMA phase
#define NCHUNKS (SLOTS / CHUNK_SLOTS) // 4

typedef float v2f __attribute__((ext_vector_type(2)));
typedef float v8f __attribute__((ext_vector_type(8)));

__global__ __launch_bounds__(THREADS)
void mlsa_cascade_kernel(const float* __restrict__ x,
                         const float* __restrict__ mc,
                         const float* __restrict__ amp,
                         const float* __restrict__ wgt,
                         float* __restrict__ out)
{
    // su[24 + s] holds u(slot s); su[0..23] is a permanent zero pad.
    __shared__ __align__(16) float su[SLOTS + MTAPS];
    // WMMA staging: interpolated coefficients for one 256-slot chunk.
    __shared__ __align__(16) float sc[CHUNK_SLOTS * NCOEF];

    const int tid  = threadIdx.x;
    const int tile = blockIdx.x;
    const int b    = blockIdx.y;

    const int s0    = 2 * tid;                   // this thread's first slot
    const int tBase = tile * TILE - HALO;        // global time of slot 0 (16-aligned)
    const int t0i   = tBase + s0;
    const int t1i   = t0i + 1;

    const float* xb  = x  + (size_t)b * T_TOTAL;
    const float* mcb = mc + (size_t)b * NFRAMES * NCOEF;

    // ---- zero pad region (slots < 0) ----
    if (tid < MTAPS / 2) {
        *(float2*)&su[2 * tid] = make_float2(0.f, 0.f);
    }

    // ---- stage-0 signal into LDS via CDNA5 async global->LDS DMA ----
    const bool v0 = (t0i >= 0) && (t0i < T_TOTAL);
    const bool v1 = (t1i >= 0) && (t1i < T_TOTAL);
    unsigned lds0 = (unsigned)(size_t)&su[MTAPS + s0];  // low 32 bits = LDS offset
    if (v0 && v1) {
        asm volatile("global_load_async_to_lds_b64 %0, %1, off"
                     :: "v"(lds0), "v"((unsigned long long)(xb + t0i)) : "memory");
    } else {
        if (v0)
            asm volatile("global_load_async_to_lds_b32 %0, %1, off"
                         :: "v"(lds0), "v"((unsigned long long)(xb + t0i)) : "memory");
        else
            su[MTAPS + s0] = 0.f;
        if (v1)
            asm volatile("global_load_async_to_lds_b32 %0, %1, off"
                         :: "v"(lds0 + 4u), "v"((unsigned long long)(xb + t1i)) : "memory");
        else
            su[MTAPS + s0 + 1] = 0.f;
    }

    // =====================================================================
    // Coefficient interpolation via V_WMMA_F32_16X16X4_F32.
    // Each 16-slot group lies inside one 80-sample frame (80 = 5*16 and the
    // tile base is 16-aligned), so per group:
    //   D(16 phases x 16 taps) = A(16x4: [1-w, w, 0, 0]) x B(4x16: [mc_n; mc_n1; 0; 0])
    // 64 groups are produced in 4 chunks of 16 (one group per wave per chunk).
    // =====================================================================
    const int wv   = tid >> 5;          // wave id 0..15
    const int lane = tid & 31;
    const int lo   = lane & 15;
    const int hi   = lane >> 4;         // 0: K rows 0/1, 1: K rows 2/3 (zero)

    float ca[MTAPS];   // coef at t0 ; index m maps to tap j = 24-m
    float cb[MTAPS];   // coef at t1
    float K0 = 0.f, K1 = 0.f;
    const int myChunk = tid >> 7;       // 128 threads consume each 256-slot chunk

#pragma unroll
    for (int c = 0; c < NCHUNKS; ++c) {
        const int g  = c * 16 + wv;           // group index 0..63
        int tg = tBase + g * 16;              // 16-aligned group start time
        int tc = tg < 0 ? 0 : (tg > T_TOTAL - 16 ? T_TOTAL - 16 : tg);
        int n  = tc / PERIOD;
        int p0 = tc - n * PERIOD;             // phase of row m=0 (p0+15 <= 79)
        int n1 = (n + 1 < NFRAMES) ? n + 1 : NFRAMES - 1;

        float wm = (float)(p0 + lo) * (1.0f / PERIOD);
        v2f af;
        af.x = hi ? 0.f : (1.0f - wm);        // A[m, K=0]  (K=2 -> 0)
        af.y = hi ? 0.f : wm;                 // A[m, K=1]  (K=3 -> 0)

        const float* mr0 = mcb + (size_t)n  * NCOEF;
        const float* mr1 = mcb + (size_t)n1 * NCOEF;

#pragma unroll
        for (int jt = 0; jt < 2; ++jt) {
            const int j = jt * 16 + lo;       // tap/coef column 0..31
            const bool jv = (hi == 0) && (j < NCOEF);
            v2f bf;
            bf.x = jv ? mr0[j] : 0.f;         // B[K=0, n]  (K=2 -> 0)
            bf.y = jv ? mr1[j] : 0.f;         // B[K=1, n]  (K=3 -> 0)

            v8f acc = {0.f, 0.f, 0.f, 0.f, 0.f, 0.f, 0.f, 0.f};
            v8f d = __builtin_amdgcn_wmma_f32_16x16x4_f32(
                        false, af, false, bf, (short)0, acc, false, false);

            if (j < NCOEF) {                  // D: lane -> (M = r + 8*hi, N = j)
                const int sb = wv * 16 + 8 * hi;   // slot-in-chunk base
#pragma unroll
                for (int r = 0; r < 8; ++r)
                    sc[(sb + r) * NCOEF + j] = d[r];
            }
        }

        __syncthreads();                      // chunk produced
        if (c == myChunk) {                   // pull my 2 slots into registers
            const int sic = s0 - c * CHUNK_SLOTS;
#pragma unroll
            for (int j = 1; j <= MTAPS; ++j) {
                ca[MTAPS - j] = sc[sic * NCOEF + j];
                cb[MTAPS - j] = sc[(sic + 1) * NCOEF + j];
            }
            K0 = expf(sc[sic * NCOEF]);
            K1 = expf(sc[(sic + 1) * NCOEF]);
        }
        __syncthreads();                      // chunk consumed, safe to reuse
    }

    // ---- wait for async DMA of x, then make the tile visible ----
    asm volatile("s_wait_asynccnt 0" ::: "memory");
    __syncthreads();

    const float a0 = amp[0];
    float y0 = a0 * su[MTAPS + s0];
    float y1 = a0 * su[MTAPS + s0 + 1];

    const bool z0 = (t0i < 0);   // reference zero-pads the signal at t < 0
    const bool z1 = (t1i < 0);

    // =====================================================================
    // 20-stage time-varying FIR cascade (VALU + LDS; WMMA loses here:
    // im2col B-matrix has only 2 useful columns -> <=12.5% utilization).
    // =====================================================================
#pragma unroll
    for (int i = 1; i <= STAGES; ++i) {
        const float wi = wgt[i];
        const float ai = amp[i];

        float win[MTAPS + 1];                 // u(prev) at slots s0-24..s0
#pragma unroll
        for (int q = 0; q < MTAPS / 2; ++q) {
            float2 v = *(const float2*)&su[s0 + 2 * q];   // 8B aligned
            win[2 * q]     = v.x;
            win[2 * q + 1] = v.y;
        }
        win[MTAPS] = su[s0 + MTAPS];

        float acc0 = 0.f, acc1 = 0.f;
#pragma unroll
        for (int m = 0; m < MTAPS; ++m) {
            acc0 = fmaf(win[m],     ca[m], acc0);   // tap j = 24-m at t0
            acc1 = fmaf(win[m + 1], cb[m], acc1);   // tap j = 24-m at t1
        }
        float u0 = z0 ? 0.f : acc0 * wi;
        float u1 = z1 ? 0.f : acc1 * wi;
        y0 = fmaf(u0, ai, y0);
        y1 = fmaf(u1, ai, y1);

        __syncthreads();                      // all reads done
        *(float2*)&su[MTAPS + s0] = make_float2(u0, u1);
        __syncthreads();                      // all writes visible
    }

    // ---- epilogue: only the 544 non-halo slots are exact ----
    if (s0 >= HALO) {
        if (t0i < T_TOTAL) out[(size_t)b * T_TOTAL + t0i] = y0 * K0;
        if (t1i < T_TOTAL) out[(size_t)b * T_TOTAL + t1i] = y1 * K1;
    }
}

extern "C" void kernel_launch(void* const* d_in, const int* in_sizes, int n_in,
                              void* d_out, int out_size, void* d_ws, size_t ws_size,
                              hipStream_t stream)
{
    const float* x   = (const float*)d_in[0];   // (B, 81920)
    const float* mcp = (const float*)d_in[1];   // (B, 1024, 25)
    const float* a   = (const float*)d_in[2];   // (21,)
    const float* wgt = (const float*)d_in[3];   // (21,)
    float* out = (float*)d_out;                 // (B, 81920)

    int B = in_sizes[0] / T_TOTAL;
    dim3 grid(NTILES, B);
    mlsa_cascade_kernel<<<grid, THREADS, 0, stream>>>(x, mcp, a, wgt, out);
}